// DynamicKGating_23940147708253
// MI455X (gfx1250) — compile-verified
//
#include <hip/hip_runtime.h>
#include <math.h>

#define B_   4
#define G_   4096
#define D_   1024
#define E_   32
#define TOK  (B_*G_)
#define CAP  160
#define KC   64
#define THRESH 0.8f

typedef __attribute__((ext_vector_type(2))) float v2f;
typedef __attribute__((ext_vector_type(8))) float v8f;

// ---------------------------------------------------------------------------
// K1: gating logits via V_WMMA_F32_16X16X4_F32, then per-token softmax +
// dynamic-k (greedy top until cumprob >= 0.8), normalized weights.
// One wave = one 16-token M tile; two N tiles (experts 0-15, 16-31).
// ---------------------------------------------------------------------------
__global__ __launch_bounds__(256) void k_gating(
    const float* __restrict__ x, const float* __restrict__ wg,
    float* __restrict__ weights, unsigned int* __restrict__ maskbits,
    float* __restrict__ gate_sum)
{
  __shared__ float sWt[E_][KC + 4];        // W chunk, transposed [expert][k]
  __shared__ float sX [8][16][KC + 4];     // per-wave x tile [row][k]
  __shared__ float sLog[8][16][E_ + 1];    // logits transpose buffer
  __shared__ float sGate[E_];              // block-level softmax-prob sums

  const int tid  = threadIdx.x;
  const int w    = tid >> 5;
  const int ln   = tid & 31;
  const int tok0 = blockIdx.x * 128;       // 128 tokens per workgroup
  const int row0 = w * 16;

  if (tid < E_) sGate[tid] = 0.0f;

  v8f c0 = {0,0,0,0,0,0,0,0};
  v8f c1 = {0,0,0,0,0,0,0,0};

  const int half = (ln >> 4) * 2;          // lanes 16-31 hold K+2/K+3 (ISA 7.12.2)
  const int r    = ln & 15;

  for (int k0 = 0; k0 < D_; k0 += KC) {
    __syncthreads();
    // stage W chunk transposed (coalesced global reads)
    for (int f = tid; f < KC * E_; f += 256) {
      int kk = f >> 5, ee = f & 31;
      sWt[ee][kk] = wg[(size_t)(k0 + kk) * E_ + ee];
    }
    // stage this wave's 16 x-rows (row-contiguous, coalesced)
    for (int rr = 0; rr < 16; ++rr) {
      const float* src = x + (size_t)(tok0 + row0 + rr) * D_ + k0 + ln * 2;
      v2f v = *(const v2f*)src;
      sX[w][rr][ln * 2]     = v.x;
      sX[w][rr][ln * 2 + 1] = v.y;
    }
    __syncthreads();

    for (int ks = 0; ks < KC; ks += 4) {
      const int kk = ks + half;
      v2f a  = *(const v2f*)&sX[w][r][kk];        // A: 16x4, lane = row M
      v2f b0 = *(const v2f*)&sWt[r][kk];          // B: 4x16, lane = col N
      v2f b1 = *(const v2f*)&sWt[r + 16][kk];
      c0 = __builtin_amdgcn_wmma_f32_16x16x4_f32(false, a, false, b0,
                                                 (short)0, c0, false, false);
      c1 = __builtin_amdgcn_wmma_f32_16x16x4_f32(false, a, false, b1,
                                                 (short)0, c1, false, false);
    }
  }

  // transpose logits: C/D layout -> sLog[wave][row][expert]
  {
    const int mrow = (ln >> 4) * 8;
    const int ncol = ln & 15;
#pragma unroll
    for (int i = 0; i < 8; ++i) {
      sLog[w][mrow + i][ncol]      = c0[i];
      sLog[w][mrow + i][16 + ncol] = c1[i];
    }
  }
  __syncthreads();

  if (tid < 128) {
    const int token = tok0 + tid;
    const int w2 = tid >> 4, r2 = tid & 15;
    float l[E_];
    float mx = -1e30f;
#pragma unroll
    for (int e = 0; e < E_; ++e) { l[e] = sLog[w2][r2][e]; mx = fmaxf(mx, l[e]); }
    float sum = 0.f;
#pragma unroll
    for (int e = 0; e < E_; ++e) { l[e] = __expf(l[e] - mx); sum += l[e]; }
    const float inv = 1.0f / sum;
#pragma unroll
    for (int e = 0; e < E_; ++e) { l[e] *= inv; atomicAdd(&sGate[e], l[e]); }

    // greedy descending pick until cumulative prob crosses threshold
    unsigned int sel = 0; float cum = 0.f;
    for (int it = 0; it < E_; ++it) {
      int best = 0; float bv = -1.f;
#pragma unroll
      for (int e = 0; e < E_; ++e) {
        float pv = ((sel >> e) & 1u) ? -1.f : l[e];
        if (pv > bv) { bv = pv; best = e; }   // strict > == stable argsort tiebreak
      }
      sel |= 1u << best; cum += bv;
      if (cum >= THRESH) break;
    }
    const float inv2 = 1.0f / cum;            // cum == sum of selected probs
#pragma unroll
    for (int e = 0; e < E_; ++e)
      weights[(size_t)token * E_ + e] = ((sel >> e) & 1u) ? l[e] * inv2 : 0.f;
    maskbits[token] = sel;
  }
  __syncthreads();
  if (tid < E_) {
    const int b = tok0 / G_;                  // whole block in one batch
    atomicAdd(&gate_sum[b * E_ + tid], sGate[tid]);
  }
}

// ---------------------------------------------------------------------------
// K2: per-(b,e) exclusive prefix count over the 4096 tokens, capacity trim.
// One wave32 per (b,e); ballot+popcount prefix; pos_e laid out [e][token].
// ---------------------------------------------------------------------------
__global__ __launch_bounds__(32) void k_scan(
    const unsigned int* __restrict__ maskbits,
    int* __restrict__ pos_e, int* __restrict__ mask_count)
{
  const int b  = blockIdx.x >> 5;
  const int e  = blockIdx.x & 31;
  const int ln = threadIdx.x;
  const unsigned int lt = (1u << ln) - 1u;
  int running = 0, kept = 0;
  for (int c = 0; c < G_ / 32; ++c) {
    const int token = b * G_ + c * 32 + ln;
    const unsigned int m = (maskbits[token] >> e) & 1u;
    const unsigned int bal = (unsigned int)__ballot((int)m);
    const int pos  = running + __popc(bal & lt);
    const int keep = (m && pos < CAP) ? 1 : 0;
    pos_e[(size_t)e * TOK + token] = keep ? pos : -1;   // coalesced
    running += __popc(bal);          // cumsum uses pre-trim mask (ref semantics)
    kept    += keep;
  }
  for (int o = 16; o > 0; o >>= 1) kept += __shfl_down(kept, o);
  if (ln == 0) mask_count[blockIdx.x] = kept;           // density numerator
}

// ---------------------------------------------------------------------------
// K2b: pos_sum[token] = sum over experts of kept positions.
// ---------------------------------------------------------------------------
__global__ __launch_bounds__(256) void k_possum(
    const int* __restrict__ pos_e, int* __restrict__ pos_sum)
{
  const int token = blockIdx.x * 256 + threadIdx.x;
  int s = 0;
#pragma unroll
  for (int e = 0; e < E_; ++e) {
    int v = pos_e[(size_t)e * TOK + token];             // coalesced per plane
    s += (v > 0) ? v : 0;
  }
  pos_sum[token] = s;
}

// ---------------------------------------------------------------------------
// K3: the bandwidth-bound writer. One wave per (token, expert) row of 160;
// float4 stores -> global_store_b128 at full line width. Writes every
// element (zeros included) so d_out is fully defined each call.
// ---------------------------------------------------------------------------
__global__ __launch_bounds__(256) void k_write(
    const float* __restrict__ weights, const int* __restrict__ pos_e,
    const int* __restrict__ pos_sum, float* __restrict__ dispatch,
    float* __restrict__ combine)
{
  const int w  = threadIdx.x >> 5, ln = threadIdx.x & 31;
  const size_t row = (size_t)blockIdx.x * 8 + w;
  const int token = (int)(row >> 5);
  const int e     = (int)(row & 31);
  const int ps    = pos_sum[token];
  const int pe    = pos_e[(size_t)e * TOK + token];
  const bool valid = (pe >= 0) && (ps < CAP);
  const float wt   = weights[row];
  float* dr = dispatch + row * CAP;
  float* cr = combine  + row * CAP;
#pragma unroll
  for (int base = 0; base < CAP; base += 128) {
    const int c = base + ln * 4;
    if (c < CAP) {
      float4 dv, cv;
      dv.x = (valid && c + 0 == ps) ? 1.0f : 0.0f;
      dv.y = (valid && c + 1 == ps) ? 1.0f : 0.0f;
      dv.z = (valid && c + 2 == ps) ? 1.0f : 0.0f;
      dv.w = (valid && c + 3 == ps) ? 1.0f : 0.0f;
      cv.x = dv.x * wt; cv.y = dv.y * wt; cv.z = dv.z * wt; cv.w = dv.w * wt;
      *(float4*)(dr + c) = dv;
      *(float4*)(cr + c) = cv;
    }
  }
}

// ---------------------------------------------------------------------------
// K4: loss = mean_{b,e}(density_proxy * density) * e^2
// ---------------------------------------------------------------------------
__global__ __launch_bounds__(128) void k_loss(
    const float* __restrict__ gate_sum, const int* __restrict__ mask_count,
    float* __restrict__ loss_out)
{
  __shared__ float red[128];
  const int t = threadIdx.x;
  const float dp = gate_sum[t] * (1.0f / G_);
  const float dn = (float)mask_count[t] * (1.0f / G_);
  red[t] = dp * dn;
  __syncthreads();
  for (int s = 64; s > 0; s >>= 1) { if (t < s) red[t] += red[t + s]; __syncthreads(); }
  if (t == 0) *loss_out = red[0] * ((float)(E_ * E_) / (float)(B_ * E_));
}

// ---------------------------------------------------------------------------
extern "C" void kernel_launch(void* const* d_in, const int* in_sizes, int n_in,
                              void* d_out, int out_size, void* d_ws, size_t ws_size,
                              hipStream_t stream)
{
  (void)in_sizes; (void)n_in; (void)out_size; (void)ws_size;
  const float* x  = (const float*)d_in[0];
  const float* wg = (const float*)d_in[1];
  float* out = (float*)d_out;

  char* ws = (char*)d_ws;
  size_t off = 0;
  float*        weights    = (float*)(ws + off); off += (size_t)TOK * E_ * sizeof(float);
  int*          pos_e      = (int*)(ws + off);   off += (size_t)TOK * E_ * sizeof(int);
  unsigned int* maskbits   = (unsigned int*)(ws + off); off += (size_t)TOK * sizeof(unsigned int);
  int*          pos_sum    = (int*)(ws + off);   off += (size_t)TOK * sizeof(int);
  float*        gate_sum   = (float*)(ws + off); off += (size_t)B_ * E_ * sizeof(float);
  int*          mask_count = (int*)(ws + off);   off += (size_t)B_ * E_ * sizeof(int);

  hipMemsetAsync(gate_sum, 0, (size_t)B_ * E_ * sizeof(float), stream);

  k_gating<<<TOK / 128, 256, 0, stream>>>(x, wg, weights, maskbits, gate_sum);
  k_scan  <<<B_ * E_,   32,  0, stream>>>(maskbits, pos_e, mask_count);
  k_possum<<<TOK / 256, 256, 0, stream>>>(pos_e, pos_sum);

  float* dispatch = out;
  float* combine  = out + (size_t)TOK * E_ * CAP;
  k_write<<<(TOK * E_) / 8, 256, 0, stream>>>(weights, pos_e, pos_sum, dispatch, combine);
  k_loss <<<1, 128, 0, stream>>>(gate_sum, mask_count, out + 2ull * TOK * E_ * CAP);
}